// RLGHGT_3092376453610
// MI455X (gfx1250) — compile-verified
//
#include <hip/hip_runtime.h>
#include <math.h>

typedef float v2f __attribute__((ext_vector_type(2)));
typedef float v8f __attribute__((ext_vector_type(8)));

#define DEV_INLINE __device__ __forceinline__

constexpr int N_  = 50000;
constexpr int E_  = 400000;
constexpr int D_  = 128;
constexpr int H_  = 4;
constexpr int DH_ = 32;
constexpr int T_  = 3;
constexpr int R_  = 6;
constexpr float EPS_ = 1e-5f;
constexpr float INV_SQRT_DH = 0.17677669529663687f; // 1/sqrt(32)

// ---------- ordered-uint encoding for float atomic max ----------
DEV_INLINE unsigned enc_f(float f) {
    unsigned u = __float_as_uint(f);
    return (u & 0x80000000u) ? ~u : (u | 0x80000000u);
}
DEV_INLINE float dec_f(unsigned e) {
    return (e & 0x80000000u) ? __uint_as_float(e & 0x7FFFFFFFu)
                             : __uint_as_float(~e);
}

// ---------- workspace init (mx/den/agg zeroed each layer) ----------
__global__ void init_ws_kernel(float* __restrict__ agg,
                               unsigned* __restrict__ mx,
                               float* __restrict__ den) {
    int i = blockIdx.x * blockDim.x + threadIdx.x;
    if (i < N_ * D_) agg[i] = 0.0f;
    if (i < N_ * H_) { mx[i] = 0u; den[i] = 0.0f; }
}

// ---------- typed linear via V_WMMA_F32_16X16X4_F32 ----------
// One wave computes a 16-node x 32-col output slab: A tile registers are
// loaded once (32 x v2f = 64 VGPRs) and reused for two 16x16 N-tiles.
// B fragments are preloaded into a distinct register array so the loads
// clause up and WMMAs issue back-to-back instead of wait-per-wmma.
// ntype is sorted -> tile spans [t0,t1]; per-row blend only at boundaries.
__global__ void __launch_bounds__(128)
proj_typed_wmma(const float* __restrict__ x,
                const int*   __restrict__ ntype,
                const float* __restrict__ W,   // [T,128,128]
                float*       __restrict__ out) {
    const int lane = threadIdx.x;                 // 0..31 (wave32)
    const int m0   = blockIdx.x * 16;             // node tile
    const int n0   = threadIdx.y * 32;            // 32-col slab per wave
    const int lo16 = lane & 15;
    const int hi   = lane >> 4;                   // 0 or 1

    const int t0 = ntype[m0];
    const int t1 = ntype[m0 + 15];

    // A 16x4 f32 frags for all 32 K-steps: lane -> M = lane%16,
    // VGPR j -> K = 4*i + 2*hi + j  (two consecutive floats -> b64 load)
    v2f afrag[32];
    #pragma unroll
    for (int i = 0; i < 32; ++i) {
        const int kb = 4 * i + 2 * hi;
        afrag[i] = *(const v2f*)(x + (size_t)(m0 + lo16) * D_ + kb);
    }

    v8f res0 = {}, res1 = {};
    for (int t = t0; t <= t1; ++t) {
        const float* Wt = W + (size_t)t * (D_ * D_);
        v8f acc0 = {}, acc1 = {};
        {   // N-tile 0: cols [n0, n0+16)
            v2f bf[32];
            #pragma unroll
            for (int i = 0; i < 32; ++i) {
                const int kb = 4 * i + 2 * hi;
                bf[i][0] = Wt[(size_t)kb       * D_ + n0 + lo16];
                bf[i][1] = Wt[(size_t)(kb + 1) * D_ + n0 + lo16];
            }
            #pragma unroll
            for (int i = 0; i < 32; ++i) {
                acc0 = __builtin_amdgcn_wmma_f32_16x16x4_f32(
                           false, afrag[i], false, bf[i], (short)0, acc0, false, false);
            }
        }
        {   // N-tile 1: cols [n0+16, n0+32)
            v2f bf[32];
            #pragma unroll
            for (int i = 0; i < 32; ++i) {
                const int kb = 4 * i + 2 * hi;
                bf[i][0] = Wt[(size_t)kb       * D_ + n0 + 16 + lo16];
                bf[i][1] = Wt[(size_t)(kb + 1) * D_ + n0 + 16 + lo16];
            }
            #pragma unroll
            for (int i = 0; i < 32; ++i) {
                acc1 = __builtin_amdgcn_wmma_f32_16x16x4_f32(
                           false, afrag[i], false, bf[i], (short)0, acc1, false, false);
            }
        }
        if (t0 != t1) {
            #pragma unroll
            for (int j = 0; j < 8; ++j) {
                int row = m0 + j + 8 * hi;       // C/D layout: VGPR j -> M = j + 8*hi
                if (ntype[row] == t) { res0[j] = acc0[j]; res1[j] = acc1[j]; }
            }
        } else {
            res0 = acc0; res1 = acc1;
        }
    }
    #pragma unroll
    for (int j = 0; j < 8; ++j) {
        int row = m0 + j + 8 * hi;
        out[(size_t)row * D_ + n0      + lo16] = res0[j];
        out[(size_t)row * D_ + n0 + 16 + lo16] = res1[j];
    }
}

// ---------- attention logits: a = (k[src]^T W_att q[dst]) * pri / sqrt(DH) ----------
__global__ void attn_logits_kernel(const float* __restrict__ kbuf,
                                   const float* __restrict__ qbuf,
                                   const int*   __restrict__ src,
                                   const int*   __restrict__ dst,
                                   const int*   __restrict__ etype,
                                   const float* __restrict__ Watt,  // [H,R,32,32]
                                   const float* __restrict__ priL,  // [H,R]
                                   float*       __restrict__ abuf,
                                   unsigned*    __restrict__ mxbuf) {
    int idx = blockIdx.x * blockDim.x + threadIdx.x;
    if (idx >= E_ * H_) return;
    int e = idx >> 2, h = idx & 3;
    int r = etype[e], s = src[e], d = dst[e];
    const float* kr = kbuf + (size_t)s * D_ + h * DH_;
    const float* qr = qbuf + (size_t)d * D_ + h * DH_;
    const float* W  = Watt + (size_t)(h * R_ + r) * (DH_ * DH_);

    float qv[DH_];
    #pragma unroll
    for (int o = 0; o < DH_; ++o) qv[o] = qr[o];

    float a = 0.0f;
    for (int dd = 0; dd < DH_; ++dd) {
        float u = 0.0f;
        #pragma unroll
        for (int o = 0; o < DH_; ++o) u += W[dd * DH_ + o] * qv[o];
        a += kr[dd] * u;
    }
    a *= priL[h * R_ + r] * INV_SQRT_DH;
    abuf[idx] = a;
    atomicMax(mxbuf + (size_t)d * H_ + h, enc_f(a));
}

// ---------- softmax numerator: ex = exp(a - mx[dst]); den += ex ----------
__global__ void softmax_num_kernel(const int*      __restrict__ dst,
                                   const unsigned* __restrict__ mxbuf,
                                   float*          __restrict__ abuf,
                                   float*          __restrict__ denbuf) {
    int idx = blockIdx.x * blockDim.x + threadIdx.x;
    if (idx >= E_ * H_) return;
    int e = idx >> 2, h = idx & 3, d = dst[e];
    float ex = expf(abuf[idx] - dec_f(mxbuf[(size_t)d * H_ + h]));
    abuf[idx] = ex;
    atomicAdd(denbuf + (size_t)d * H_ + h, ex);
}

// ---------- aggregate: agg[dst] += att * (v[src]^T W_msg)  (mv recomputed, never stored) ----------
__global__ void aggregate_kernel(const float* __restrict__ vbuf,
                                 const int*   __restrict__ src,
                                 const int*   __restrict__ dst,
                                 const int*   __restrict__ etype,
                                 const float* __restrict__ Wmsg,  // [H,R,32,32]
                                 const float* __restrict__ abuf,
                                 const float* __restrict__ denbuf,
                                 float*       __restrict__ aggbuf) {
    int idx = blockIdx.x * blockDim.x + threadIdx.x;
    if (idx >= E_ * H_) return;
    int e = idx >> 2, h = idx & 3;
    int r = etype[e], s = src[e], d = dst[e];
    float att = abuf[idx] / denbuf[(size_t)d * H_ + h];
    const float* vr = vbuf + (size_t)s * D_ + h * DH_;
    const float* W  = Wmsg + (size_t)(h * R_ + r) * (DH_ * DH_);

    float vv[DH_];
    #pragma unroll
    for (int dd = 0; dd < DH_; ++dd) vv[dd] = vr[dd];

    for (int o = 0; o < DH_; ++o) {
        float m = 0.0f;
        #pragma unroll
        for (int dd = 0; dd < DH_; ++dd) m += vv[dd] * W[dd * DH_ + o];
        atomicAdd(aggbuf + (size_t)d * D_ + h * DH_ + o, att * m);
    }
}

// ---------- skip-gate + residual + LayerNorm (one wave per node) ----------
__global__ void skip_ln_kernel(const float* __restrict__ x,
                               const float* __restrict__ ha,
                               const int*   __restrict__ ntype,
                               const float* __restrict__ skipL, // [T]
                               const float* __restrict__ g,
                               const float* __restrict__ b,
                               float*       __restrict__ xout) {
    int lane = threadIdx.x & 31;
    int node = blockIdx.x * (blockDim.x >> 5) + (threadIdx.x >> 5);
    if (node >= N_) return;
    float sk = 1.0f / (1.0f + expf(-skipL[ntype[node]]));
    float z[4];
    float s = 0.0f;
    #pragma unroll
    for (int i = 0; i < 4; ++i) {
        int c = i * 32 + lane;
        float xv = x[(size_t)node * D_ + c];
        float hv = ha[(size_t)node * D_ + c];
        z[i] = xv + (hv * sk + xv * (1.0f - sk));
        s += z[i];
    }
    #pragma unroll
    for (int o = 16; o; o >>= 1) s += __shfl_xor(s, o, 32);
    float mu = s * (1.0f / 128.0f);
    float v = 0.0f;
    #pragma unroll
    for (int i = 0; i < 4; ++i) { float t = z[i] - mu; v += t * t; }
    #pragma unroll
    for (int o = 16; o; o >>= 1) v += __shfl_xor(v, o, 32);
    float rs = rsqrtf(v * (1.0f / 128.0f) + EPS_);
    #pragma unroll
    for (int i = 0; i < 4; ++i) {
        int c = i * 32 + lane;
        xout[(size_t)node * D_ + c] = (z[i] - mu) * rs * g[c] + b[c];
    }
}

// ---------- final: softmax(agg_w)-weighted mix + LayerNorm ----------
__global__ void final_mix_kernel(const float* __restrict__ X1,
                                 const float* __restrict__ X2,
                                 const float* __restrict__ agg_w, // [2]
                                 const float* __restrict__ g,
                                 const float* __restrict__ b,
                                 float*       __restrict__ out) {
    int lane = threadIdx.x & 31;
    int node = blockIdx.x * (blockDim.x >> 5) + (threadIdx.x >> 5);
    if (node >= N_) return;
    float w0a = agg_w[0], w1a = agg_w[1];
    float mw = fmaxf(w0a, w1a);
    float e0 = expf(w0a - mw), e1 = expf(w1a - mw);
    float inv = 1.0f / (e0 + e1);
    float w0 = e0 * inv, w1 = e1 * inv;

    float z[4];
    float s = 0.0f;
    #pragma unroll
    for (int i = 0; i < 4; ++i) {
        int c = i * 32 + lane;
        z[i] = w0 * X1[(size_t)node * D_ + c] + w1 * X2[(size_t)node * D_ + c];
        s += z[i];
    }
    #pragma unroll
    for (int o = 16; o; o >>= 1) s += __shfl_xor(s, o, 32);
    float mu = s * (1.0f / 128.0f);
    float v = 0.0f;
    #pragma unroll
    for (int i = 0; i < 4; ++i) { float t = z[i] - mu; v += t * t; }
    #pragma unroll
    for (int o = 16; o; o >>= 1) v += __shfl_xor(v, o, 32);
    float rs = rsqrtf(v * (1.0f / 128.0f) + EPS_);
    #pragma unroll
    for (int i = 0; i < 4; ++i) {
        int c = i * 32 + lane;
        out[(size_t)node * D_ + c] = (z[i] - mu) * rs * g[c] + b[c];
    }
}

extern "C" void kernel_launch(void* const* d_in, const int* in_sizes, int n_in,
                              void* d_out, int out_size, void* d_ws, size_t ws_size,
                              hipStream_t stream) {
    const float* h     = (const float*)d_in[0];
    const int*   src   = (const int*)  d_in[1];
    const int*   dst   = (const int*)  d_in[2];
    const int*   ntype = (const int*)  d_in[3];
    const int*   etype = (const int*)  d_in[4];
    const float* Wk    = (const float*)d_in[5];
    const float* Wq    = (const float*)d_in[6];
    const float* Wv    = (const float*)d_in[7];
    const float* Wa    = (const float*)d_in[8];
    const float* Watt  = (const float*)d_in[9];
    const float* Wmsg  = (const float*)d_in[10];
    const float* pri   = (const float*)d_in[11];
    const float* skip  = (const float*)d_in[12];
    const float* ln_g  = (const float*)d_in[13];
    const float* ln_b  = (const float*)d_in[14];
    const float* agg_w = (const float*)d_in[15];
    const float* agg_g = (const float*)d_in[16];
    const float* agg_b = (const float*)d_in[17];

    // workspace carving (floats): 7*N*D + E*H + 2*N*H  ~= 187 MB
    float* ws      = (float*)d_ws;
    float* kbuf    = ws;
    float* qbuf    = kbuf   + (size_t)N_ * D_;
    float* vbuf    = qbuf   + (size_t)N_ * D_;
    float* habuf   = vbuf   + (size_t)N_ * D_;
    float* aggbuf  = habuf  + (size_t)N_ * D_;
    float* X1      = aggbuf + (size_t)N_ * D_;
    float* X2      = X1     + (size_t)N_ * D_;
    float* abuf    = X2     + (size_t)N_ * D_;
    unsigned* mxbuf = (unsigned*)(abuf + (size_t)E_ * H_);
    float* denbuf  = (float*)(mxbuf + (size_t)N_ * H_);

    dim3 projGrid(N_ / 16);                  // 3125 node tiles
    dim3 projBlk(32, 4);                     // 4 waves, one 16x32 slab each
    int ehBlocks  = (E_ * H_ + 255) / 256;   // 6250
    int initBlocks= (N_ * D_ + 255) / 256;   // 25000
    int lnBlocks  = N_ / 4;                  // 12500 (blockDim 128 = 4 waves = 4 nodes)

    const float* x = h;
    for (int l = 0; l < 2; ++l) {
        const float* WkL   = Wk   + (size_t)l * T_ * D_ * D_;
        const float* WqL   = Wq   + (size_t)l * T_ * D_ * D_;
        const float* WvL   = Wv   + (size_t)l * T_ * D_ * D_;
        const float* WaL   = Wa   + (size_t)l * T_ * D_ * D_;
        const float* WattL = Watt + (size_t)l * H_ * R_ * DH_ * DH_;
        const float* WmsgL = Wmsg + (size_t)l * H_ * R_ * DH_ * DH_;
        const float* priL  = pri  + (size_t)l * H_ * R_;
        const float* skipL = skip + (size_t)l * T_;
        const float* gL    = ln_g + (size_t)l * D_;
        const float* bL    = ln_b + (size_t)l * D_;
        float* xout = (l == 0) ? X1 : X2;

        init_ws_kernel<<<initBlocks, 256, 0, stream>>>(aggbuf, mxbuf, denbuf);
        proj_typed_wmma<<<projGrid, projBlk, 0, stream>>>(x, ntype, WkL, kbuf);
        proj_typed_wmma<<<projGrid, projBlk, 0, stream>>>(x, ntype, WqL, qbuf);
        proj_typed_wmma<<<projGrid, projBlk, 0, stream>>>(x, ntype, WvL, vbuf);
        attn_logits_kernel<<<ehBlocks, 256, 0, stream>>>(kbuf, qbuf, src, dst, etype,
                                                         WattL, priL, abuf, mxbuf);
        softmax_num_kernel<<<ehBlocks, 256, 0, stream>>>(dst, mxbuf, abuf, denbuf);
        aggregate_kernel<<<ehBlocks, 256, 0, stream>>>(vbuf, src, dst, etype,
                                                       WmsgL, abuf, denbuf, aggbuf);
        proj_typed_wmma<<<projGrid, projBlk, 0, stream>>>(aggbuf, ntype, WaL, habuf);
        skip_ln_kernel<<<lnBlocks, 128, 0, stream>>>(x, habuf, ntype, skipL, gL, bL, xout);
        x = xout;
    }
    final_mix_kernel<<<lnBlocks, 128, 0, stream>>>(X1, X2, agg_w, agg_g, agg_b, (float*)d_out);
}